// GroupedExpertMLPFast_69234872811782
// MI455X (gfx1250) — compile-verified
//
#include <hip/hip_runtime.h>
#include <hip/hip_bf16.h>

#define N_EXPERTS 8
#define D_MODEL   512
#define D_FF      1024
#define T_MAX     256
#define CHUNK     128
#define XS_STRIDE (D_MODEL + 4)   // pad to avoid 16-way LDS bank conflicts
#define HS_STRIDE (CHUNK + 4)

typedef float v2f __attribute__((ext_vector_type(2)));
typedef float v8f __attribute__((ext_vector_type(8)));

// -------- Routing: build per-expert token lists (deterministic serial scan) --------
// ws int layout: [0..7] counts, [8 .. 8+8*T_MAX) token lists
__global__ void moe_route_kernel(const int* __restrict__ eids, int* __restrict__ ws_i, int T) {
    if (threadIdx.x == 0) {
        int cnt[N_EXPERTS];
        for (int e = 0; e < N_EXPERTS; ++e) cnt[e] = 0;
        for (int t = 0; t < T; ++t) {
            int e = eids[t] & (N_EXPERTS - 1);
            ws_i[N_EXPERTS + e * T_MAX + cnt[e]] = t;
            cnt[e]++;
        }
        for (int e = 0; e < N_EXPERTS; ++e) ws_i[e] = cnt[e];
    }
}

// -------- Fused expert MLP: 16-token tile of one expert per block --------
__global__ __launch_bounds__(256) void moe_expert_mlp_kernel(
    const float* __restrict__ x,
    const float* __restrict__ w1,
    const float* __restrict__ w3,
    const float* __restrict__ w2,
    const int*   __restrict__ ws_i,
    float*       __restrict__ out)
{
    __shared__ float xs[16 * XS_STRIDE];   // x tile [16 tokens][512], padded
    __shared__ float hs[16 * HS_STRIDE];   // hidden chunk [16 tokens][128], padded
    __shared__ int   toks[16];

    const int e    = blockIdx.x;
    const int cnt  = ws_i[e];
    const int t0   = blockIdx.y * 16;
    if (t0 >= cnt) return;                       // uniform per block: whole waves exit
    const int nval = min(16, cnt - t0);

    const int tid   = threadIdx.x;
    const int lane  = tid & 31;
    const int wave  = tid >> 5;
    const int halfk = lane >> 4;                 // 0 or 1 (K / M-high selector)
    const int lm    = lane & 15;                 // 0..15 (M for A, N for B/D)

    if (tid < 16) toks[tid] = (tid < nval) ? ws_i[N_EXPERTS + e * T_MAX + t0 + tid] : -1;
    __syncthreads();

    // Load x tile into LDS, zero-filling invalid token rows. 16 threads/row, 8 float4 each.
    {
        const int row  = tid >> 4;
        const int cidx = tid & 15;
        const int tok  = toks[row];
        const float* src = (tok >= 0) ? (x + (size_t)tok * D_MODEL) : nullptr;
        float* dst = xs + row * XS_STRIDE;
        #pragma unroll
        for (int i = 0; i < 8; ++i) {
            int c = (cidx + i * 16) * 4;
            float4 v = make_float4(0.f, 0.f, 0.f, 0.f);
            if (src) v = *(const float4*)(src + c);
            *(float4*)(dst + c) = v;
        }
    }
    __syncthreads();

    const float* w1e = w1 + (size_t)e * D_FF * D_MODEL;
    const float* w3e = w3 + (size_t)e * D_FF * D_MODEL;
    const float* w2e = w2 + (size_t)e * D_MODEL * D_FF;

    // Output accumulators: wave owns 4 d-tiles (16 tokens x 64 of d_model)
    v8f acc[4];
    #pragma unroll
    for (int j = 0; j < 4; ++j) acc[j] = v8f{};

    for (int f0 = 0; f0 < D_FF; f0 += CHUNK) {
        // ---- GEMM1: this wave computes f-slice [f0 + wave*16, +16) of gate & up ----
        {
            const int fbase = f0 + wave * 16;
            const float* w1row = w1e + (size_t)(fbase + lm) * D_MODEL + halfk * 2;
            const float* w3row = w3e + (size_t)(fbase + lm) * D_MODEL + halfk * 2;
            const float* xrow  = xs  + lm * XS_STRIDE + halfk * 2;
            v8f g = v8f{}, u = v8f{};
            for (int k = 0; k < D_MODEL; k += 4) {
                v2f a  = { xrow[k],  xrow[k + 1]  };
                v2f b1 = { w1row[k], w1row[k + 1] };
                v2f b3 = { w3row[k], w3row[k + 1] };
                g = __builtin_amdgcn_wmma_f32_16x16x4_f32(false, a, false, b1, (short)0, g, false, false);
                u = __builtin_amdgcn_wmma_f32_16x16x4_f32(false, a, false, b3, (short)0, u, false, false);
            }
            // SwiGLU -> hidden chunk in LDS. D layout: VGPR r holds M=r+8*halfk, N=lm.
            #pragma unroll
            for (int r = 0; r < 8; ++r) {
                float gv = g[r];
                float h  = (gv / (1.0f + __expf(-gv))) * u[r];
                hs[(r + 8 * halfk) * HS_STRIDE + wave * 16 + lm] = h;
            }
        }
        __syncthreads();

        // ---- GEMM2 partial: acc[j] += hidden_chunk x w2^T for wave's 4 d-tiles ----
        const float* hrow = hs + lm * HS_STRIDE + halfk * 2;
        #pragma unroll
        for (int j = 0; j < 4; ++j) {
            const int d0 = (wave * 4 + j) * 16;
            const float* w2row = w2e + (size_t)(d0 + lm) * D_FF + f0 + halfk * 2;
            v8f c = acc[j];
            for (int k = 0; k < CHUNK; k += 4) {
                v2f a = { hrow[k],  hrow[k + 1]  };
                v2f b = { w2row[k], w2row[k + 1] };
                c = __builtin_amdgcn_wmma_f32_16x16x4_f32(false, a, false, b, (short)0, c, false, false);
            }
            acc[j] = c;
        }
        __syncthreads();
    }

    // ---- Scatter output rows back to tokens ----
    #pragma unroll
    for (int j = 0; j < 4; ++j) {
        const int d0 = (wave * 4 + j) * 16;
        #pragma unroll
        for (int r = 0; r < 8; ++r) {
            const int m = r + 8 * halfk;
            if (m < nval) {
                out[(size_t)toks[m] * D_MODEL + d0 + lm] = acc[j][r];
            }
        }
    }
}

extern "C" void kernel_launch(void* const* d_in, const int* in_sizes, int n_in,
                              void* d_out, int out_size, void* d_ws, size_t ws_size,
                              hipStream_t stream) {
    const float* x    = (const float*)d_in[0];
    const int*   eids = (const int*)  d_in[1];
    const float* w1   = (const float*)d_in[2];
    const float* w3   = (const float*)d_in[3];
    const float* w2   = (const float*)d_in[4];
    float*       out  = (float*)d_out;
    const int T = in_sizes[1];

    int* ws_i = (int*)d_ws;

    moe_route_kernel<<<1, 32, 0, stream>>>(eids, ws_i, T);

    dim3 grid(N_EXPERTS, (T + 15) / 16);
    moe_expert_mlp_kernel<<<grid, 256, 0, stream>>>(x, w1, w3, w2, ws_i, out);
}